// HmmNet_635655160381
// MI455X (gfx1250) — compile-verified
//
#include <hip/hip_runtime.h>
#include <cstdint>

// ---------------- problem constants (from reference) ----------------
#define B_SZ    256
#define T_LEN   1024
#define TP1     1025
#define NB_     32
#define NA_     16

// ---------------- pipeline configuration ----------------
#define CHUNK   8                     // timesteps per TDM chunk
#define PIPE    4                     // 4-deep chunk pipeline (~86KB LDS, CDNA5 has 320KB/WGP)
#define NCHUNK  (T_LEN / CHUNK)       // 128
// action tile LDS padding: TDM inserts 2 pad dwords every 16 data dwords
// -> per-t row: 512 data + 64 pad = 576 dwords; per-state stride = 18 dwords
// (18 is conflict-free over 64 banks for 32 lanes)
#define ALP_ROW 576
#define NB_STRIDE 18

typedef unsigned int u32;
typedef unsigned long long u64;
typedef u32 __attribute__((ext_vector_type(4))) u32x4;
typedef int __attribute__((ext_vector_type(4))) i32x4;
typedef int __attribute__((ext_vector_type(8))) i32x8;
typedef float __attribute__((ext_vector_type(8))) f32x8;

#if __has_builtin(__builtin_amdgcn_tensor_load_to_lds)
#define HAVE_TDM 1
#else
#define HAVE_TDM 0
#endif

#if __has_builtin(__builtin_amdgcn_s_wait_tensorcnt) && HAVE_TDM
#define WAIT_TENSOR(n) __builtin_amdgcn_s_wait_tensorcnt(n)
#else
#define WAIT_TENSOR(n)
#endif

// Opaque barrier: makes TDM-written LDS contents "visible" to the compiler
// (no undef-folding / CSE of LDS loads, no hoisting above the tensor wait)
// while keeping the loads themselves plain ds_load ops.
#define MEM_BARRIER() __asm__ __volatile__("" ::: "memory")

// Before reading chunk n from LDS: allow PIPE-2 newer chunks (4 TDM ops each)
// to remain in flight; near the tail fewer chunks are outstanding.
#define WAIT_FOR_CHUNK(n) do {                      \
    const int _p = (NCHUNK - 1) - (n);              \
    if (_p >= PIPE - 2)      { WAIT_TENSOR(4 * (PIPE - 2)); } \
    else if (_p == 1)        { WAIT_TENSOR(4); }    \
    else                     { WAIT_TENSOR(0); }    \
    MEM_BARRIER();                                  \
  } while (0)

#if HAVE_TDM
// 1-D contiguous TDM copy: n_elem dwords from global -> LDS byte offset,
// optional LDS-side padding (pad_int/pad_amt are the encoded D# field codes).
// D# packing per CDNA5 ISA §8.3 (group0) / §8.4 (group1).
static __device__ __forceinline__ void tdm_copy(u32 lds_off, const void* gptr,
                                                u32 n_elem, u32 pad_en,
                                                u32 pad_int, u32 pad_amt) {
  const u64 ga = (u64)(uintptr_t)gptr;
  u32x4 g0;
  g0.x = 0x1u;                                   // count=1 (valid), no gather
  g0.y = lds_off;                                // lds_addr [63:32]
  g0.z = (u32)ga;                                // global_addr [95:64]
  g0.w = (u32)((ga >> 32) & 0x01FFFFFFull)       // global_addr [120:96]
         | (2u << 30);                           // type=2 ("image")
  i32x8 g1;
  g1[0] = (int)((2u << 16)                       // data_size = 4 bytes
                | (pad_en << 20)
                | (pad_int << 22)
                | (pad_amt << 25));
  g1[1] = (int)((n_elem & 0xFFFFu) << 16);       // tensor_dim0[15:0]
  g1[2] = (int)((n_elem >> 16) | (1u << 16));    // tensor_dim0[31:16] | tensor_dim1=1
  g1[3] = (int)((n_elem & 0xFFFFu) << 16);       // tensor_dim1 hi=0 | tile_dim0=n
  g1[4] = 1;                                     // tile_dim1=1, tile_dim2=0
  g1[5] = (int)n_elem;                           // tensor_dim0_stride lo32
  g1[6] = 0;                                     // stride hi16 | dim1_stride lo16
  g1[7] = 0;
  i32x4 z4 = {0, 0, 0, 0};
#if defined(__clang_major__) && (__clang_major__ >= 23)
  i32x8 z8 = {0, 0, 0, 0, 0, 0, 0, 0};
  __builtin_amdgcn_tensor_load_to_lds(g0, g1, z4, z4, z8, 0);
#else
  __builtin_amdgcn_tensor_load_to_lds(g0, g1, z4, z4, 0);
#endif
}
#else
// Fallback: synchronous lane-strided copy reproducing the padded LDS layout.
static __device__ __forceinline__ void lds_copy(u32* dst, const u32* src,
                                                u32 n, u32 pad_en, int lane) {
  for (u32 i = (u32)lane; i < n; i += 32u) {
    u32 di = pad_en ? (i + 2u * (i >> 4)) : i;   // 2 pad dwords / 16 dwords
    dst[di] = src[i];
  }
}
#endif

static __device__ __forceinline__ float wave_max32(float x) {
#pragma unroll
  for (int i = 16; i > 0; i >>= 1) x = fmaxf(x, __shfl_xor(x, i, 32));
  return x;
}
static __device__ __forceinline__ float wave_sum32(float x) {
#pragma unroll
  for (int i = 16; i > 0; i >>= 1) x += __shfl_xor(x, i, 32);
  return x;
}

// One wave (32 lanes) per batch row; lane = HMM state nb.
// Scan runs in LINEAR space: g = exp(f - C_exp*ln2), renormalized once per
// chunk by an exact power of two (exponent of max(g) accumulated in C_exp).
// Per-chunk operands are TDM-streamed to LDS, then double-buffered in
// ext-vector REGISTERS so LDS latency overlaps the previous chunk's math.
__global__ __launch_bounds__(32) void hmm_fwd_kernel(
    const float* __restrict__ alp,   // (B, 1025, 32, 16)
    const float* __restrict__ stp,   // (B, 1025, 32, 2)
    const float* __restrict__ str,   // (B, 1025, 32)
    const int*   __restrict__ act,   // (B, 1024)
    const int*   __restrict__ len_,  // (B,)
    float*       __restrict__ partial)
{
  __shared__ float s_alp [PIPE][CHUNK * ALP_ROW];
  __shared__ float s_stop[PIPE][CHUNK * 64];
  __shared__ float s_strt[PIPE][CHUNK * 32];
  __shared__ int   s_act [PIPE][CHUNK];

  const int b     = blockIdx.x;
  const int lane  = threadIdx.x;
  const int len   = len_[b];           // uniform
  const int tstop = len - 1;           // in [0, 1023]

#if HAVE_TDM
  // Flat LDS address: low 32 bits == LDS byte offset (ISA §10.2 aperture map)
  const u32 off_alp  = (u32)(uintptr_t)&s_alp[0][0];
  const u32 off_stop = (u32)(uintptr_t)&s_stop[0][0];
  const u32 off_strt = (u32)(uintptr_t)&s_strt[0][0];
  const u32 off_act  = (u32)(uintptr_t)&s_act[0][0];
#endif

  auto issue = [&](int c) {
    const int bufi = c & (PIPE - 1);
    const size_t base = (size_t)b * TP1 + (size_t)(c * CHUNK);
#if HAVE_TDM
    tdm_copy(off_alp  + (u32)bufi * (CHUNK * ALP_ROW * 4u),
             alp + base * 512u, CHUNK * 512u, /*pad*/1u, /*16 dw*/3u, /*2 dw*/1u);
    tdm_copy(off_stop + (u32)bufi * (CHUNK * 64u * 4u),
             stp + base * 64u,  CHUNK * 64u,  0u, 0u, 0u);
    tdm_copy(off_strt + (u32)bufi * (CHUNK * 32u * 4u),
             str + base * 32u,  CHUNK * 32u,  0u, 0u, 0u);
    tdm_copy(off_act  + (u32)bufi * (CHUNK * 4u),
             act + (size_t)b * T_LEN + (size_t)(c * CHUNK), CHUNK, 0u, 0u, 0u);
#else
    lds_copy((u32*)&s_alp[bufi][0],  (const u32*)(alp + base * 512u),
             CHUNK * 512u, 1u, lane);
    lds_copy((u32*)&s_stop[bufi][0], (const u32*)(stp + base * 64u),
             CHUNK * 64u, 0u, lane);
    lds_copy((u32*)&s_strt[bufi][0], (const u32*)(str + base * 32u),
             CHUNK * 32u, 0u, lane);
    lds_copy((u32*)&s_act[bufi][0],
             (const u32*)(act + (size_t)b * T_LEN + (size_t)(c * CHUNK)),
             CHUNK, 0u, lane);
#endif
  };

  // Register operand buffers: ext-vectors -> insert/extractelement only,
  // guaranteed VGPR-resident (no alloca, no scratch spills).
  f32x8 rA_alp, rA_bet, rA_omb, rA_st;
  f32x8 rB_alp, rB_bet, rB_omb, rB_st;

  // Plain (non-volatile) LDS reads: provenance-clear -> ds_load; adjacent
  // beta/omb pairs merge into ds_load_b64; ordering pinned by MEM_BARRIER().
  auto preload = [&](int c, f32x8& ra, f32x8& rb, f32x8& ro, f32x8& rs) {
    const int bufi = c & (PIPE - 1);
    i32x8 a;
#pragma unroll
    for (int tt = 0; tt < CHUNK; ++tt)
      a[tt] = s_act[bufi][tt];                     // broadcast reads
#pragma unroll
    for (int tt = 0; tt < CHUNK; ++tt) {
      ra[tt] = s_alp [bufi][tt * ALP_ROW + lane * NB_STRIDE + a[tt]];
      rb[tt] = s_stop[bufi][tt * 64 + lane * 2 + 0];
      ro[tt] = s_stop[bufi][tt * 64 + lane * 2 + 1];
      rs[tt] = s_strt[bufi][tt * 32 + lane];
    }
  };

  float g     = 0.0f;   // linear-space state, max(g) kept near [1,2)
  int   C_exp = 0;      // accumulated power-of-2 renormalization exponent
  float g_cap = 1.0f;   // snapshot at t == len-1
  int   c_cap = 0;

  auto compute = [&](int c, const f32x8& ra, const f32x8& rb,
                     const f32x8& ro, const f32x8& rs, bool first) {
#pragma unroll
    for (int tt = 0; tt < CHUNK; ++tt) {
      // Off-chain transcendentals & products:
      //   g' = (g*e^omb + s*e^start)*e^a  ==  fma(s, e^(start+a), g*e^(omb+a))
      const float ea  = __expf(ra[tt]);
      const float eb  = __expf(rb[tt]);
      const float eoa = __expf(ro[tt]) * ea;
      const float esa = __expf(rs[tt]) * ea;
      if (first && tt == 0) {
        g = esa;                                   // exp(start0 + a_lp0)
      } else {
        const float s = wave_sum32(g * eb);        // single 5-stage butterfly
        g = __builtin_fmaf(s, esa, g * eoa);       // one fma after the butterfly
      }
      const int t = c * CHUNK + tt;
      if (t == tstop) { g_cap = g; c_cap = C_exp; }
    }
    // Exact power-of-2 renormalization once per chunk: keeps max(g) in ~[1,2).
    const float m  = wave_max32(g);
    const u32  mb  = __float_as_uint(m);
    const int  ee  = (int)((mb >> 23) & 0xFFu);            // biased exponent
    C_exp += ee - 127;
    g *= __uint_as_float((u32)(254 - ee) << 23);           // exact 2^-(ee-127)
  };

  // ---- prologue: fill TDM pipeline, preload chunks 0 and 1 into registers ----
#pragma unroll
  for (int c = 0; c < PIPE; ++c) issue(c);
  WAIT_FOR_CHUNK(0); preload(0, rA_alp, rA_bet, rA_omb, rA_st);
  WAIT_FOR_CHUNK(1); preload(1, rB_alp, rB_bet, rB_omb, rB_st);
  compute(0, rA_alp, rA_bet, rA_omb, rA_st, true);
  issue(PIPE);

  // ---- steady state: compute chunk c while chunk c+1's LDS->reg loads fly ----
  for (int c = 1; c < NCHUNK; c += 2) {
    if (c + 1 < NCHUNK) { WAIT_FOR_CHUNK(c + 1);
                          preload(c + 1, rA_alp, rA_bet, rA_omb, rA_st); }
    compute(c, rB_alp, rB_bet, rB_omb, rB_st, false);
    if (c + PIPE < NCHUNK) issue(c + PIPE);

    if (c + 1 < NCHUNK) {
      if (c + 2 < NCHUNK) { WAIT_FOR_CHUNK(c + 2);
                            preload(c + 2, rB_alp, rB_bet, rB_omb, rB_st); }
      compute(c + 1, rA_alp, rA_bet, rA_omb, rA_st, false);
      if (c + 1 + PIPE < NCHUNK) issue(c + 1 + PIPE);
    }
  }

  // x0 = log f at t=len-1 ; x1 = stop_logps[b, len, :, STOP_IX]
  const float x0 = (float)c_cap * 0.6931471805599453f + __logf(g_cap);
  const float x1 = stp[((size_t)b * TP1 + (size_t)len) * 64u + (size_t)(lane * 2)];
  const float v  = x0 + x1;
  const float m  = wave_max32(v);
  const float s  = wave_sum32(__expf(v - m));
  if (lane == 0) partial[b] = m + __logf(s);
}

// Deterministic fixed-order reduction: out[0] = -sum_b partial[b]
__global__ __launch_bounds__(256) void neg_sum_kernel(const float* __restrict__ partial,
                                                      float* __restrict__ out) {
  __shared__ float sb[8];
  const int tid = threadIdx.x;
  float v = partial[tid];
  v = wave_sum32(v);
  if ((tid & 31) == 0) sb[tid >> 5] = v;
  __syncthreads();
  if (tid == 0) {
    float s = 0.0f;
#pragma unroll
    for (int i = 0; i < 8; ++i) s += sb[i];
    out[0] = -s;
  }
}

extern "C" void kernel_launch(void* const* d_in, const int* in_sizes, int n_in,
                              void* d_out, int out_size, void* d_ws, size_t ws_size,
                              hipStream_t stream) {
  (void)in_sizes; (void)n_in; (void)out_size; (void)ws_size;
  const float* alp  = (const float*)d_in[0];   // action_logps
  const float* stp  = (const float*)d_in[1];   // stop_logps
  const float* str  = (const float*)d_in[2];   // start_logps
  const int*   act  = (const int*)d_in[3];     // actions
  const int*   len_ = (const int*)d_in[4];     // lengths
  float* partial = (float*)d_ws;               // 256 floats of scratch
  float* out     = (float*)d_out;

  hmm_fwd_kernel<<<dim3(B_SZ), dim3(32), 0, stream>>>(alp, stp, str, act, len_, partial);
  neg_sum_kernel<<<dim3(1), dim3(256), 0, stream>>>(partial, out);
}